// SpatialMaxout2d_49160195670149
// MI455X (gfx1250) — compile-verified
//
#include <hip/hip_runtime.h>
#include <stdint.h>

// SpatialMaxout2d (kh=kw=2) on x:(32,64,224,224) f32.
// H,W are multiples of 2 -> padding windows are fully cropped, so the op
// reduces to: per 2x2 window, pass the (first) argmax element through, scale
// the other 3 by alpha[p]. Pure streaming: ~822MB traffic -> HBM-bound
// (~35us at 23.3 TB/s). Data path: CDNA5 async global->LDS b128 loads,
// 4-deep pipelined with s_wait_asynccnt, b128 global stores.

#define TPB   256
#define ITERS 8     // units per thread
#define PIPE  4     // async pipeline depth (buffers)

#if defined(__has_builtin)
#if __has_builtin(__builtin_amdgcn_global_load_async_to_lds_b128)
#define USE_ASYNC_LDS 1
#endif
#endif
#ifndef USE_ASYNC_LDS
#define USE_ASYNC_LDS 0
#endif

#if __has_builtin(__builtin_amdgcn_s_wait_asynccnt)
#define WAIT_ASYNC(n) __builtin_amdgcn_s_wait_asynccnt(n)
#else
#define WAIT_ASYNC(n) asm volatile("s_wait_asynccnt %0" ::"i"(n) : "memory")
#endif

typedef int v4i __attribute__((ext_vector_type(4)));
typedef __attribute__((address_space(1))) v4i* g_v4i_ptr;   // global (AS1)
typedef __attribute__((address_space(3))) v4i* l_v4i_ptr;   // LDS (AS3)

// One 2x2 window: out[p] = v[p] if p is the first argmax else alpha[p]*v[p].
__device__ __forceinline__ void blend_win(float v0, float v1, float v2, float v3,
                                          float a0, float a1, float a2, float a3,
                                          float& o0, float& o1, float& o2, float& o3)
{
    float m  = fmaxf(fmaxf(v0, v1), fmaxf(v2, v3));
    bool  p0 = (v0 == m);
    bool  p1 = (v1 == m) && !p0;
    bool  p2 = (v2 == m) && !(p0 | p1);
    bool  p3 = !(p0 | p1 | p2);
    o0 = p0 ? v0 : a0 * v0;
    o1 = p1 ? v1 : a1 * v1;
    o2 = p2 ? v2 : a2 * v2;
    o3 = p3 ? v3 : a3 * v3;
}

// Work unit u -> float offset of top float4. Chunk = 2 contiguous rows (448 f),
// 56 float4 columns per chunk; bottom row is +224 floats.
__device__ __forceinline__ size_t unit_goff(unsigned u)
{
    unsigned chunk = u / 56u;
    unsigned col   = u - chunk * 56u;
    return (size_t)chunk * 448u + (size_t)col * 4u;
}

#if USE_ASYNC_LDS
__device__ __forceinline__ void issue_pair(const float* __restrict__ x, unsigned u,
                                           unsigned U, float4* tslot, float4* bslot)
{
    // Clamp (don't skip): keeps ASYNCcnt bookkeeping exact for the whole wave.
    unsigned uc = u < U ? u : U - 1u;
    size_t   g  = unit_goff(uc);
    // Global source pointers (AS1, int4-typed as the builtin requires;
    // const dropped via integer round-trip — loads don't write).
    g_v4i_ptr gt = (g_v4i_ptr)(uintptr_t)(x + g);
    g_v4i_ptr gb = (g_v4i_ptr)(uintptr_t)(x + g + 224);
    // Low 32 bits of a flat shared-memory address == LDS byte offset (ISA 10.2).
    l_v4i_ptr lt = (l_v4i_ptr)(uint32_t)(uintptr_t)tslot;
    l_v4i_ptr lb = (l_v4i_ptr)(uint32_t)(uintptr_t)bslot;
    __builtin_amdgcn_global_load_async_to_lds_b128(gt, lt, 0, 0);
    __builtin_amdgcn_global_load_async_to_lds_b128(gb, lb, 0, 0);
}
#endif

__global__ __launch_bounds__(TPB) void spatial_maxout_2x2(
    const float* __restrict__ x, const float* __restrict__ alpha,
    float* __restrict__ out, unsigned U)
{
    const float a0 = alpha[0], a1 = alpha[1], a2 = alpha[2], a3 = alpha[3];
    const unsigned tid    = threadIdx.x;
    const unsigned base_u = blockIdx.x * (TPB * ITERS) + tid;

#if USE_ASYNC_LDS
    // [buffer][top/bottom][thread] : 32 KB, 4-deep pipeline.
    __shared__ float4 smem[PIPE][2][TPB];

    // Prologue: issue PIPE-1 = 3 iterations ahead (6 loads in flight / wave).
#pragma unroll
    for (int p = 0; p < PIPE - 1 && p < ITERS; ++p)
        issue_pair(x, base_u + (unsigned)p * TPB, U, &smem[p][0][tid], &smem[p][1][tid]);

#pragma unroll
    for (int i = 0; i < ITERS; ++i) {
        const unsigned u   = base_u + (unsigned)i * TPB;
        const int      buf = i & (PIPE - 1);
        // Issue PIPE-1 ahead; steady-state keeps 6 b128 loads outstanding.
        if (i + PIPE - 1 < ITERS) {
            const int nbuf = (i + PIPE - 1) & (PIPE - 1);  // == (i-1)&3: read last iter
            issue_pair(x, u + (PIPE - 1) * TPB, U, &smem[nbuf][0][tid], &smem[nbuf][1][tid]);
            WAIT_ASYNC(2 * (PIPE - 1));   // 6: this buffer's 2 loads are done
        } else if (i + 3 == ITERS) {
            WAIT_ASYNC(4);
        } else if (i + 2 == ITERS) {
            WAIT_ASYNC(2);
        } else {
            WAIT_ASYNC(0);
        }
        float4 tv = smem[buf][0][tid];
        float4 bv = smem[buf][1][tid];
        // Retire LDS reads before this buffer is re-targeted next iteration.
        asm volatile("s_wait_dscnt 0" ::: "memory");
        float4 to, bo;
        blend_win(tv.x, tv.y, bv.x, bv.y, a0, a1, a2, a3, to.x, to.y, bo.x, bo.y);
        blend_win(tv.z, tv.w, bv.z, bv.w, a0, a1, a2, a3, to.z, to.w, bo.z, bo.w);
        if (u < U) {
            size_t g = unit_goff(u);
            *reinterpret_cast<float4*>(out + g)       = to;
            *reinterpret_cast<float4*>(out + g + 224) = bo;
        }
    }
#else
    // Fallback: plain b128 streaming + gfx1250 global_prefetch_b8.
#pragma unroll
    for (int i = 0; i < ITERS; ++i) {
        const unsigned u  = base_u + (unsigned)i * TPB;
        const unsigned uc = u < U ? u : U - 1u;
        const size_t   g  = unit_goff(uc);
        const float4   tv = *reinterpret_cast<const float4*>(x + g);
        const float4   bv = *reinterpret_cast<const float4*>(x + g + 224);
        __builtin_prefetch((const void*)(x + g + (size_t)448 * 64), 0, 0);
        float4 to, bo;
        blend_win(tv.x, tv.y, bv.x, bv.y, a0, a1, a2, a3, to.x, to.y, bo.x, bo.y);
        blend_win(tv.z, tv.w, bv.z, bv.w, a0, a1, a2, a3, to.z, to.w, bo.z, bo.w);
        if (u < U) {
            *reinterpret_cast<float4*>(out + g)       = to;
            *reinterpret_cast<float4*>(out + g + 224) = bo;
        }
    }
#endif
}

extern "C" void kernel_launch(void* const* d_in, const int* in_sizes, int n_in,
                              void* d_out, int out_size, void* d_ws, size_t ws_size,
                              hipStream_t stream)
{
    const float* x     = (const float*)d_in[0];
    const float* alpha = (const float*)d_in[1];
    float*       out   = (float*)d_out;

    const long long N      = (long long)in_sizes[0];      // 32*64*224*224
    const unsigned  planes = (unsigned)(N / (224 * 224)); // B*C = 2048
    const unsigned  chunks = planes * 112u;               // row pairs
    const unsigned  U      = chunks * 56u;                // float4 columns

    const unsigned perBlock = TPB * ITERS;
    const unsigned blocks   = (U + perBlock - 1) / perBlock; // 6272

    spatial_maxout_2x2<<<blocks, TPB, 0, stream>>>(x, alpha, out, U);
}